// PSRoIPool_82016695485109
// MI455X (gfx1250) — compile-verified
//
#include <hip/hip_runtime.h>
#include <hip/hip_bf16.h>

typedef __attribute__((ext_vector_type(2))) float v2f;
typedef __attribute__((ext_vector_type(8))) float v8f;

#define GG   1029        // C = G*G*D = 49*21
#define FH   128
#define FW   128
#define SC   0.0625f

// One wave32 handles 16 output bins. WMMA f32 16x16x4 with B=ones is a bank of
// 16 independent f32 accumulators: D[m][*] = sum_k A[m][k] + C[m][*]. Lane L
// feeds row M = L%16 (lanes 0-15 -> K{0,1}, lanes 16-31 -> K{2,3}). The body is
// unrolled 2x so 4 gathered loads are in flight before the loadcnt drain,
// doubling per-wave MLP for this L2-resident gather.
__global__ __launch_bounds__(256) void psroi_wmma_kernel(
    const float* __restrict__ features, const float* __restrict__ rois,
    float* __restrict__ out, long total)
{
    const int  lane   = threadIdx.x & 31;
    const int  wid    = threadIdx.x >> 5;
    const long base16 = ((long)blockIdx.x * 8 + wid) * 16;
    const long bin    = base16 + (lane & 15);   // lanes L and L+16 share bin L%16

    int cnt = 0, hs = 0, ws0 = 0, w = 1;
    const float* fp = features;

    if (bin < total) {
        int r   = (int)(bin / GG);
        int c   = (int)(bin - (long)r * GG);    // channel == bin % 1029
        int rem = c % 49;
        int ph  = rem / 7;
        int pw  = rem - ph * 7;

        const float* roi = rois + r * 5;
        int   bi = (int)roi[0];
        // jnp.round = round-half-even -> rintf
        float xs = rintf(roi[1]) * SC;
        float ys = rintf(roi[2]) * SC;
        float xe = rintf(roi[3] + 1.0f) * SC;
        float ye = rintf(roi[4] + 1.0f) * SC;
        float rw = fmaxf(xe - xs, 0.1f);
        float rh = fmaxf(ye - ys, 0.1f);
        float bh = rh / 7.0f;
        float bw = rw / 7.0f;
        // non-contracted mul+add to match reference (separate mul / add HLOs)
        float hsf = fminf(fmaxf(floorf(__fadd_rn(__fmul_rn((float)ph,      bh), ys)), 0.f), (float)FH);
        float hef = fminf(fmaxf(ceilf (__fadd_rn(__fmul_rn((float)(ph + 1), bh), ys)), 0.f), (float)FH);
        float wsf = fminf(fmaxf(floorf(__fadd_rn(__fmul_rn((float)pw,      bw), xs)), 0.f), (float)FW);
        float wef = fminf(fmaxf(ceilf (__fadd_rn(__fmul_rn((float)(pw + 1), bw), xs)), 0.f), (float)FW);
        hs  = (int)hsf;  int he = (int)hef;
        ws0 = (int)wsf;  int we = (int)wef;
        int hh = he - hs;
        w   = we - ws0;
        cnt = hh * w;
        if (w <= 0) { w = 1; cnt = 0; }
        fp = features + ((long)bi * GG + c) * (long)(FH * FW);
    }

    // Wave-uniform trip count (EXEC must be all-ones for every WMMA).
    int mx = cnt;
    #pragma unroll
    for (int off = 16; off > 0; off >>= 1) {
        int o = __shfl_xor(mx, off, 32);
        mx = (o > mx) ? o : mx;
    }
    const int pairs = (mx + 7) >> 3;      // 8 elements per bin per outer iter

    const int klo   = (lane >> 4) << 1;   // lanes 0-15 feed K{0,1}; 16-31 feed K{2,3}
    const v2f bones = {1.0f, 1.0f};       // B = ones(4x16)
    v8f acc = {};

    for (int p = 0; p < pairs; ++p) {
        // ---- gather phase: 4 independent loads issued before any drain ----
        float v0 = 0.0f, v1 = 0.0f, v2 = 0.0f, v3 = 0.0f;
        int   i0 = (p << 3) + klo;        // first element of K-slice 0
        if (i0 < cnt) {
            int q   = i0 / w;
            int rmd = i0 - q * w;
            v0 = fp[(hs + q) * FW + (ws0 + rmd)];
            if (i0 + 1 < cnt) {
                ++rmd; if (rmd == w) { rmd = 0; ++q; }
                v1 = fp[(hs + q) * FW + (ws0 + rmd)];
            }
        }
        int i4 = i0 + 4;                  // first element of K-slice 1
        if (i4 < cnt) {
            int q   = i4 / w;
            int rmd = i4 - q * w;
            v2 = fp[(hs + q) * FW + (ws0 + rmd)];
            if (i4 + 1 < cnt) {
                ++rmd; if (rmd == w) { rmd = 0; ++q; }
                v3 = fp[(hs + q) * FW + (ws0 + rmd)];
            }
        }
        // ---- reduce phase: two chained WMMAs into the same accumulators ----
        v2f a01 = {v0, v1};
        v2f a23 = {v2, v3};
        acc = __builtin_amdgcn_wmma_f32_16x16x4_f32(
            false, a01, false, bones, (short)0, acc, false, false);
        acc = __builtin_amdgcn_wmma_f32_16x16x4_f32(
            false, a23, false, bones, (short)0, acc, false, false);
    }

    // C/D layout: VGPR v -> lanes 0-15 hold M=v, lanes 16-31 hold M=v+8,
    // every N column identical. Lane 0 exports bins 0-7, lane 16 bins 8-15.
    __shared__ float ssum[8][16];
    if (lane == 0) {
        #pragma unroll
        for (int v = 0; v < 8; ++v) ssum[wid][v] = acc[v];
    } else if (lane == 16) {
        #pragma unroll
        for (int v = 0; v < 8; ++v) ssum[wid][8 + v] = acc[v];
    }
    __syncthreads();

    if (lane < 16 && bin < total) {
        float s = ssum[wid][lane];
        out[bin] = (cnt > 0) ? (s / (float)cnt) : 0.0f;   // coalesced 64B store per wave
    }
}

extern "C" void kernel_launch(void* const* d_in, const int* in_sizes, int n_in,
                              void* d_out, int out_size, void* d_ws, size_t ws_size,
                              hipStream_t stream) {
    const float* features = (const float*)d_in[0];   // [N, 1029, 128, 128] f32
    const float* rois     = (const float*)d_in[1];   // [R, 5] f32
    float*       out      = (float*)d_out;           // [R, 21, 7, 7] f32

    const int  R     = in_sizes[1] / 5;
    const long total = (long)R * GG;                 // == out_size
    const int  binsPerBlock = 128;                   // 8 waves * 16 bins
    const int  blocks = (int)((total + binsPerBlock - 1) / binsPerBlock);

    psroi_wmma_kernel<<<blocks, 256, 0, stream>>>(features, rois, out, total);
}